// ConvBlock_82721070121701
// MI455X (gfx1250) — compile-verified
//
#include <hip/hip_runtime.h>
#include <hip/hip_bf16.h>

#define NNODES 100000
#define FDIM   128
#define FDIM2  256
#define NEDGE  800000
#define BN_EPS 1e-5f

typedef float __attribute__((ext_vector_type(2))) v2f;
typedef float __attribute__((ext_vector_type(4))) v4f;
typedef float __attribute__((ext_vector_type(8))) v8f;

__device__ __forceinline__ void atomAddF(float* p, float v) {
    unsafeAtomicAdd(p, v);   // global_atomic_add_f32
}

// ---------------- zero small stats region ----------------
__global__ void k_zero(float* __restrict__ p, int n) {
    int i = blockIdx.x * blockDim.x + threadIdx.x;
    if (i < n) p[i] = 0.f;
}

// ---------------- agg = (1+eps0) * x0 ----------------
__global__ void k_init_agg(const float* __restrict__ x0, const float* __restrict__ eps0,
                           float* __restrict__ agg) {
    int i = blockIdx.x * blockDim.x + threadIdx.x;     // v4 element index
    const int total = NNODES * FDIM / 4;
    if (i >= total) return;
    v4f xv = ((const v4f*)x0)[i];
    v4f ev = ((const v4f*)eps0)[i & 31];               // 32 v4-chunks per row
    ((v4f*)agg)[i] = (1.0f + ev) * xv;
}

// ---------------- edge scatter: agg[dst] += (1+eps_h) * relu(x[src] (+eattr)) ----
// one wave (32 lanes) per edge; lane covers 4 features
__global__ void k_edge(const float* __restrict__ x, const int* __restrict__ ei,
                       const float* __restrict__ eattr, const float* __restrict__ epsrow,
                       float* __restrict__ agg) {
    int widx = (blockIdx.x * blockDim.x + threadIdx.x) >> 5;
    if (widx >= NEDGE) return;
    int lane = threadIdx.x & 31;
    int src = ei[widx];
    int dst = ei[NEDGE + widx];
    int f = lane * 4;
    v4f m = *(const v4f*)(x + (size_t)src * FDIM + f);
    if (eattr) {   // streaming 410MB: non-temporal
        v4f ev = __builtin_nontemporal_load((const v4f*)(eattr + (size_t)widx * FDIM + f));
        m += ev;
    }
    v4f s = 1.0f + *(const v4f*)(epsrow + f);
    float* ap = agg + (size_t)dst * FDIM + f;
    atomAddF(ap + 0, fmaxf(m.x, 0.f) * s.x);
    atomAddF(ap + 1, fmaxf(m.y, 0.f) * s.y);
    atomAddF(ap + 2, fmaxf(m.z, 0.f) * s.z);
    atomAddF(ap + 3, fmaxf(m.w, 0.f) * s.w);
}

// ---------------- GEMM1: h1 = agg[N,128] @ W1[128,256] + b1 ; stats1 += col sums ----
// wave per 16x16 tile; fp32 WMMA 16x16x4
__global__ void k_gemm1(const float* __restrict__ A, const float* __restrict__ W,
                        const float* __restrict__ bias, float* __restrict__ out,
                        float* __restrict__ stats) {
    const int lane = threadIdx.x & 31;
    const int wave = threadIdx.x >> 5;
    const int row0 = blockIdx.x * 16;
    const int col0 = (blockIdx.y * 8 + wave) * 16;
    const int m  = lane & 15;
    const int hi = lane >> 4;
    const int kb = hi * 2;
    const float* arow = A + (size_t)(row0 + m) * FDIM + kb;
    const float* wcol = W + (size_t)kb * FDIM2 + col0 + m;
    v8f c = {};
    #pragma unroll
    for (int k = 0; k < FDIM; k += 4) {
        v2f a = *(const v2f*)(arow + k);
        v2f b;
        b.x = wcol[(size_t)k * FDIM2];
        b.y = wcol[(size_t)(k + 1) * FDIM2];
        c = __builtin_amdgcn_wmma_f32_16x16x4_f32(false, a, false, b, (short)0, c, false, false);
    }
    const int col = col0 + m;
    const float bv = bias[col];
    float s1 = 0.f, s2 = 0.f;
    #pragma unroll
    for (int v = 0; v < 8; ++v) {
        float val = c[v] + bv;
        out[(size_t)(row0 + hi * 8 + v) * FDIM2 + col] = val;
        s1 += val;
        s2 += val * val;
    }
    s1 += __shfl_xor(s1, 16);
    s2 += __shfl_xor(s2, 16);
    if (hi == 0) {
        atomAddF(&stats[col], s1);
        atomAddF(&stats[FDIM2 + col], s2);
    }
}

// ---------------- GEMM2: h2 = relu(bn1(h1)) @ W2[256,128] + b2 ; stats2 ----
__global__ void k_gemm2(const float* __restrict__ H, const float* __restrict__ sc,
                        const float* __restrict__ sh, const float* __restrict__ W,
                        const float* __restrict__ bias, float* __restrict__ out,
                        float* __restrict__ stats) {
    const int lane = threadIdx.x & 31;
    const int wave = threadIdx.x >> 5;
    const int row0 = blockIdx.x * 16;
    const int col0 = wave * 16;                    // 8 waves cover 128 cols
    const int m  = lane & 15;
    const int hi = lane >> 4;
    const int kb = hi * 2;
    const float* arow = H + (size_t)(row0 + m) * FDIM2 + kb;
    const float* wcol = W + (size_t)kb * FDIM + col0 + m;
    v8f c = {};
    #pragma unroll 8
    for (int k = 0; k < FDIM2; k += 4) {
        v2f a;
        a.x = fmaxf(arow[k]     * sc[k + kb]     + sh[k + kb],     0.f);
        a.y = fmaxf(arow[k + 1] * sc[k + kb + 1] + sh[k + kb + 1], 0.f);
        v2f b;
        b.x = wcol[(size_t)k * FDIM];
        b.y = wcol[(size_t)(k + 1) * FDIM];
        c = __builtin_amdgcn_wmma_f32_16x16x4_f32(false, a, false, b, (short)0, c, false, false);
    }
    const int col = col0 + m;
    const float bv = bias[col];
    float s1 = 0.f, s2 = 0.f;
    #pragma unroll
    for (int v = 0; v < 8; ++v) {
        float val = c[v] + bv;
        out[(size_t)(row0 + hi * 8 + v) * FDIM + col] = val;
        s1 += val;
        s2 += val * val;
    }
    s1 += __shfl_xor(s1, 16);
    s2 += __shfl_xor(s2, 16);
    if (hi == 0) {
        atomAddF(&stats[col], s1);
        atomAddF(&stats[FDIM + col], s2);
    }
}

// ---------------- BN finalize: scale = g*rsqrt(var+eps), shift = b - mean*scale ----
__global__ void k_bn_fin(const float* __restrict__ stats, const float* __restrict__ g,
                         const float* __restrict__ b, float* __restrict__ scale,
                         float* __restrict__ shift, int nf, float invN) {
    int d = blockIdx.x * blockDim.x + threadIdx.x;
    if (d >= nf) return;
    float mean = stats[d] * invN;
    float var  = stats[nf + d] * invN - mean * mean;
    float sc   = g[d] * rsqrtf(var + BN_EPS);
    scale[d] = sc;
    shift[d] = b[d] - mean * sc;
}

// ---------------- stats of relu(bn2(h2_pre)) per column ----------------
__global__ void k_stats3(const float* __restrict__ h2, const float* __restrict__ sc2,
                         const float* __restrict__ sh2, float* __restrict__ stats) {
    int d = threadIdx.x;                 // 128 columns
    float sc = sc2[d], sh = sh2[d];
    float s1 = 0.f, s2 = 0.f;
    for (int r = blockIdx.x; r < NNODES; r += gridDim.x) {
        float v = fmaxf(sc * h2[(size_t)r * FDIM + d] + sh, 0.f);
        s1 += v;
        s2 += v * v;
    }
    atomAddF(&stats[d], s1);
    atomAddF(&stats[FDIM + d], s2);
}

// ---------------- final: out = relu(bn3(relu(bn2(h2_pre)))) in place ----------------
__global__ void k_final(float* __restrict__ out, const float* __restrict__ sc2,
                        const float* __restrict__ sh2, const float* __restrict__ sc3,
                        const float* __restrict__ sh3) {
    int i = blockIdx.x * blockDim.x + threadIdx.x;     // v4 element index
    const int total = NNODES * FDIM / 4;
    if (i >= total) return;
    int d4 = i & 31;
    v4f h  = ((v4f*)out)[i];
    v4f a2 = ((const v4f*)sc2)[d4], b2 = ((const v4f*)sh2)[d4];
    v4f a3 = ((const v4f*)sc3)[d4], b3 = ((const v4f*)sh3)[d4];
    #pragma unroll
    for (int j = 0; j < 4; ++j) {
        float v = fmaxf(a2[j] * h[j] + b2[j], 0.f);
        h[j] = fmaxf(a3[j] * v + b3[j], 0.f);
    }
    ((v4f*)out)[i] = h;
}

extern "C" void kernel_launch(void* const* d_in, const int* in_sizes, int n_in,
                              void* d_out, int out_size, void* d_ws, size_t ws_size,
                              hipStream_t stream) {
    const float* x[4]  = {(const float*)d_in[0], (const float*)d_in[1],
                          (const float*)d_in[2], (const float*)d_in[3]};
    const float* eattr = (const float*)d_in[4];
    const int*   e[4]  = {(const int*)d_in[5], (const int*)d_in[6],
                          (const int*)d_in[7], (const int*)d_in[8]};
    const float* W1    = (const float*)d_in[9];
    const float* b1    = (const float*)d_in[10];
    const float* bn1g  = (const float*)d_in[11];
    const float* bn1b  = (const float*)d_in[12];
    const float* W2    = (const float*)d_in[13];
    const float* b2    = (const float*)d_in[14];
    const float* bn2g  = (const float*)d_in[15];
    const float* bn2b  = (const float*)d_in[16];
    const float* bn3g  = (const float*)d_in[17];
    const float* bn3b  = (const float*)d_in[18];
    const float* epsp  = (const float*)d_in[19];     // [5,128]
    float* out = (float*)d_out;                      // doubles as h2_pre

    // workspace layout
    float* agg    = (float*)d_ws;                    // N*128
    float* h1     = agg + (size_t)NNODES * FDIM;     // N*256
    float* smallb = h1 + (size_t)NNODES * FDIM2;
    float* stats1 = smallb;            // 512
    float* sc1    = stats1 + 512;      // 256
    float* sh1    = sc1 + 256;         // 256
    float* stats2 = sh1 + 256;         // 256
    float* sc2    = stats2 + 256;      // 128
    float* sh2    = sc2 + 128;         // 128
    float* stats3 = sh2 + 128;         // 256
    float* sc3    = stats3 + 256;      // 128
    float* sh3    = sc3 + 128;         // 128
    const int nsmall = 512 + 256 + 256 + 256 + 128 + 128 + 256 + 128 + 128;

    const float invN = 1.0f / (float)NNODES;

    k_zero<<<(nsmall + 255) / 256, 256, 0, stream>>>(smallb, nsmall);

    k_init_agg<<<(NNODES * FDIM / 4 + 255) / 256, 256, 0, stream>>>(x[0], epsp, agg);

    for (int hop = 0; hop < 4; ++hop) {
        k_edge<<<NEDGE / 8, 256, 0, stream>>>(x[hop], e[hop],
                                              hop == 0 ? eattr : nullptr,
                                              epsp + (size_t)(hop + 1) * FDIM, agg);
    }

    k_gemm1<<<dim3(NNODES / 16, 2), 256, 0, stream>>>(agg, W1, b1, h1, stats1);
    k_bn_fin<<<1, 256, 0, stream>>>(stats1, bn1g, bn1b, sc1, sh1, FDIM2, invN);

    k_gemm2<<<dim3(NNODES / 16, 1), 256, 0, stream>>>(h1, sc1, sh1, W2, b2, out, stats2);
    k_bn_fin<<<1, 128, 0, stream>>>(stats2, bn2g, bn2b, sc2, sh2, FDIM, invN);

    k_stats3<<<512, 128, 0, stream>>>(out, sc2, sh2, stats3);
    k_bn_fin<<<1, 128, 0, stream>>>(stats3, bn3g, bn3b, sc3, sh3, FDIM, invN);

    k_final<<<(NNODES * FDIM / 4 + 255) / 256, 256, 0, stream>>>(out, sc2, sh2, sc3, sh3);
}